// Loss_40467181863079
// MI455X (gfx1250) — compile-verified
//
#include <hip/hip_runtime.h>

typedef float v2f __attribute__((ext_vector_type(2)));
typedef float v8f __attribute__((ext_vector_type(8)));

static constexpr int Bn = 4096;   // number of anchors
static constexpr int Dd = 1024;   // feature dim
static constexpr int Nn = 8192;   // samples = 2*Bn (interleaved anchor/positive)

// ------------------------------------------------------------------
// Per-row inverse L2 norms for the interleaved `samples` matrix.
// One wave (32 lanes) per row; float4 loads; shuffle reduce.
// ------------------------------------------------------------------
__global__ __launch_bounds__(256) void norms_kernel(const float* __restrict__ anc,
                                                    const float* __restrict__ pos,
                                                    float* __restrict__ invn) {
  const int wave = (blockIdx.x * blockDim.x + threadIdx.x) >> 5;
  const int lane = threadIdx.x & 31;
  const float* row = (wave & 1) ? (pos + (size_t)(wave >> 1) * Dd)
                                : (anc + (size_t)(wave >> 1) * Dd);
  float s = 0.0f;
#pragma unroll
  for (int t = 0; t < Dd / 128; ++t) {
    const float4 v = *(const float4*)(row + t * 128 + lane * 4);
    s += v.x * v.x + v.y * v.y + v.z * v.z + v.w * v.w;
  }
#pragma unroll
  for (int off = 16; off >= 1; off >>= 1) s += __shfl_xor(s, off, 32);
  if (lane == 0) invn[wave] = 1.0f / sqrtf(s);
}

__global__ void zero_kernel(float* __restrict__ p, int n) {
  const int i = blockIdx.x * blockDim.x + threadIdx.x;
  if (i < n) p[i] = 0.0f;
}

// 16 WMMAs updating the 4x4 grid of 16x16 accumulator tiles.
#define WMMA_TILE(AF, BF)                                                      \
  _Pragma("unroll")                                                            \
  for (int sm = 0; sm < 4; ++sm) {                                             \
    _Pragma("unroll")                                                          \
    for (int sn = 0; sn < 4; ++sn) {                                           \
      acc[sm][sn] = __builtin_amdgcn_wmma_f32_16x16x4_f32(                     \
          false, AF[sm], false, BF[sn], (short)0, acc[sm][sn], false, false);  \
    }                                                                          \
  }

// ------------------------------------------------------------------
// Fused GEMM (S = anchors @ samples^T, scaled by inverse norms) +
// T = 1 - acos(clip(S))/pi + masked row reduction.
// One wave computes a 64x64 tile of S via 4x4 blocks of
// v_wmma_f32_16x16x4_f32; K loop unrolled by 2 with ping-pong
// fragment register sets (branch-free, copy-free double buffering).
// ------------------------------------------------------------------
__global__ __launch_bounds__(256) void gemm_tsum_kernel(const float* __restrict__ anc,
                                                        const float* __restrict__ pos,
                                                        const float* __restrict__ invn,
                                                        float* __restrict__ rowsum,
                                                        float* __restrict__ numv) {
  const int lane  = threadIdx.x & 31;
  const int gwave = (blockIdx.x * blockDim.x + threadIdx.x) >> 5;
  const int tilesN = Nn / 64;                 // 128
  const int tM = gwave / tilesN;
  const int tN = gwave % tilesN;
  const int m0 = tM * 64;
  const int n0 = tN * 64;

  const int lrow = lane & 15;                 // row/col index within a 16-wide frag
  const int koff = (lane >> 4) * 2;           // upper half-lanes hold K+2,K+3

  // A: anchors rows (16x4 frag per sub-tile).  B[k][n] = samples[n][k].
  const float* arow[4];
  const float* brow[4];
#pragma unroll
  for (int s = 0; s < 4; ++s) {
    arow[s] = anc + (size_t)(m0 + s * 16 + lrow) * Dd;
    const int j = n0 + s * 16 + lrow;
    brow[s] = ((j & 1) ? pos : anc) + (size_t)(j >> 1) * Dd;
  }

  v8f acc[4][4];
#pragma unroll
  for (int sm = 0; sm < 4; ++sm)
#pragma unroll
    for (int sn = 0; sn < 4; ++sn)
#pragma unroll
      for (int r = 0; r < 8; ++r) acc[sm][sn][r] = 0.0f;

  v2f a0[4], b0[4], a1[4], b1[4];
#pragma unroll
  for (int s = 0; s < 4; ++s) {
    a0[s] = *(const v2f*)(arow[s] + koff);
    b0[s] = *(const v2f*)(brow[s] + koff);
  }

  // Main loop: k = 0, 8, ..., Dd-16. Processes K-chunks k and k+4,
  // always prefetching the next two chunks. No branches, no reg copies.
  for (int k = 0; k < Dd - 8; k += 8) {
#pragma unroll
    for (int s = 0; s < 4; ++s) {
      a1[s] = *(const v2f*)(arow[s] + k + 4 + koff);
      b1[s] = *(const v2f*)(brow[s] + k + 4 + koff);
    }
    WMMA_TILE(a0, b0)
#pragma unroll
    for (int s = 0; s < 4; ++s) {
      a0[s] = *(const v2f*)(arow[s] + k + 8 + koff);
      b0[s] = *(const v2f*)(brow[s] + k + 8 + koff);
    }
    WMMA_TILE(a1, b1)
  }
  // Tail: chunks Dd-8 (already in set0) and Dd-4.
#pragma unroll
  for (int s = 0; s < 4; ++s) {
    a1[s] = *(const v2f*)(arow[s] + Dd - 4 + koff);
    b1[s] = *(const v2f*)(brow[s] + Dd - 4 + koff);
  }
  WMMA_TILE(a0, b0)
  WMMA_TILE(a1, b1)

  // Epilogue: scale, clamp, T = 1 - acos(s)/pi, masked reductions.
  // C/D layout: N = lane&15; M = r + 8*(lane>>4) within each 16-row sub-tile.
  const float invPi = 0.31830988618379067154f;
#pragma unroll
  for (int sm = 0; sm < 4; ++sm) {
    const int mbase = m0 + sm * 16 + 8 * (lane >> 4);
    float ia[8];
#pragma unroll
    for (int r = 0; r < 8; ++r) ia[r] = invn[2 * (mbase + r)];
    float racc[8];
#pragma unroll
    for (int r = 0; r < 8; ++r) racc[r] = 0.0f;
#pragma unroll
    for (int sn = 0; sn < 4; ++sn) {
      const int j  = n0 + sn * 16 + (lane & 15);
      const float ij = invn[j];
#pragma unroll
      for (int r = 0; r < 8; ++r) {
        const int i = mbase + r;
        float s = acc[sm][sn][r] * ia[r] * ij;
        s = fminf(1.0f, fmaxf(-1.0f, s));
        const float t = 1.0f - acosf(s) * invPi;
        if (j == 2 * i + 1) {
          numv[i] = t;                 // exactly one lane globally owns (i, 2i+1)
        } else if (j != 2 * i) {
          racc[r] += t;                // neg contributions only
        }
      }
    }
    // Reduce across the 16 lanes of each half (columns of this tile).
#pragma unroll
    for (int off = 1; off < 16; off <<= 1)
#pragma unroll
      for (int r = 0; r < 8; ++r) racc[r] += __shfl_xor(racc[r], off, 32);
    if ((lane & 15) == 0) {
#pragma unroll
      for (int r = 0; r < 8; ++r) atomicAdd(&rowsum[mbase + r], racc[r]);
    }
  }
}

// ------------------------------------------------------------------
// loss = -log( sum_i num_i/(neg_i + num_i) / B )
// ------------------------------------------------------------------
__global__ __launch_bounds__(256) void finalize_kernel(const float* __restrict__ rowsum,
                                                       const float* __restrict__ numv,
                                                       float* __restrict__ out) {
  __shared__ float sw[8];
  float s = 0.0f;
  for (int i = threadIdx.x; i < Bn; i += 256) {
    const float n = numv[i];
    s += n / (rowsum[i] + n);
  }
#pragma unroll
  for (int off = 16; off >= 1; off >>= 1) s += __shfl_xor(s, off, 32);
  if ((threadIdx.x & 31) == 0) sw[threadIdx.x >> 5] = s;
  __syncthreads();
  if (threadIdx.x == 0) {
    float tot = 0.0f;
#pragma unroll
    for (int w = 0; w < 8; ++w) tot += sw[w];
    out[0] = -logf(tot / (float)Bn);
  }
}

extern "C" void kernel_launch(void* const* d_in, const int* in_sizes, int n_in,
                              void* d_out, int out_size, void* d_ws, size_t ws_size,
                              hipStream_t stream) {
  const float* pos = (const float*)d_in[0];   // hid_positive
  const float* anc = (const float*)d_in[1];   // hid_anchor

  float* invn   = (float*)d_ws;               // 8192 floats
  float* rowsum = invn + Nn;                  // 4096 floats (neg sums)
  float* numv   = rowsum + Bn;                // 4096 floats (pos terms)
  float* out    = (float*)d_out;

  zero_kernel<<<(2 * Bn + 255) / 256, 256, 0, stream>>>(rowsum, 2 * Bn);
  norms_kernel<<<Nn / 8, 256, 0, stream>>>(anc, pos, invn);
  // (4096/64) * (8192/64) = 8192 wave-tiles, 8 waves per 256-thread block
  gemm_tsum_kernel<<<(Bn / 64) * (Nn / 64) / 8, 256, 0, stream>>>(anc, pos, invn,
                                                                  rowsum, numv);
  finalize_kernel<<<1, 256, 0, stream>>>(rowsum, numv, out);
}